// GCNLayer_18683107737862
// MI455X (gfx1250) — compile-verified
//
#include <hip/hip_runtime.h>
#include <hip/hip_bf16.h>

#define CH 128  // IN == OUT == 128

typedef __attribute__((ext_vector_type(2))) float v2f;
typedef __attribute__((ext_vector_type(8))) float v8f;

// ---------------- degree / norm ----------------

__global__ void gcn_deg_init(float* __restrict__ deg, int n) {
    int i = blockIdx.x * blockDim.x + threadIdx.x;
    if (i < n) deg[i] = 1.0f;  // self-loop contributes 1 to every node's degree
}

__global__ void gcn_deg_accum(const int* __restrict__ dst, float* __restrict__ deg, int E) {
    int e = blockIdx.x * blockDim.x + threadIdx.x;
    if (e < E) atomicAdd(&deg[dst[e]], 1.0f);
}

__global__ void gcn_dinv(float* __restrict__ deg, int n) {
    int i = blockIdx.x * blockDim.x + threadIdx.x;
    if (i < n) deg[i] = rsqrtf(deg[i]);  // deg >= 1 always (self loops)
}

// ---------------- dense GEMM: xw = x @ W via V_WMMA_F32_16X16X4_F32 ----------------
// 8 waves / block; block = one 16-row stripe, wave w = 16-col tile w.
// A 16x4 f32 fragment: lanes 0-15 hold K = k0+{0,1}; lanes 16-31 hold K = k0+{2,3}.
// B 4x16 f32 fragment: VGPR0 = row k (lanes 0-15) / row k+2 (lanes 16-31); VGPR1 = k+1 / k+3.
// C/D 16x16 f32: VGPR r -> M = r + 8*half, N = lane&15.

__global__ __launch_bounds__(256) void gcn_gemm_wmma(
    const float* __restrict__ X, const float* __restrict__ W,
    float* __restrict__ XW, int n)
{
    const int wave = threadIdx.x >> 5;   // 0..7 -> column tile
    const int lane = threadIdx.x & 31;
    const int half = lane >> 4;
    const int l16  = lane & 15;
    const int m0   = blockIdx.x * 16;
    const int n0   = wave * 16;

    int arow = m0 + l16;
    if (arow >= n) arow = n - 1;         // clamp (duplicate row; stores are guarded)
    const float* Ap = X + (size_t)arow * CH;

    v8f c = {};
#pragma unroll
    for (int k0 = 0; k0 < CH; k0 += 4) {
        const int ka = k0 + 2 * half;
        v2f a, b;
        a.x = Ap[ka];
        a.y = Ap[ka + 1];
        b.x = W[(size_t)ka       * CH + n0 + l16];
        b.y = W[(size_t)(ka + 1) * CH + n0 + l16];
        c = __builtin_amdgcn_wmma_f32_16x16x4_f32(
                /*neg_a=*/false, a, /*neg_b=*/false, b,
                /*c_mod=*/(short)0, c, /*reuse_a=*/false, /*reuse_b=*/false);
    }

#pragma unroll
    for (int r = 0; r < 8; ++r) {
        int row = m0 + r + 8 * half;
        if (row < n) XW[(size_t)row * CH + n0 + l16] = c[r];
    }
}

// ---------------- out init: self-loop + bias (also zero-bases the atomics) ----------------

__global__ void gcn_init_out(const float* __restrict__ xw, const float* __restrict__ dinv,
                             const float* __restrict__ bias, float* __restrict__ out, int total) {
    int idx = blockIdx.x * blockDim.x + threadIdx.x;
    if (idx >= total) return;
    int i = idx >> 7;        // node
    int c = idx & (CH - 1);  // channel
    float di = dinv[i];
    out[idx] = bias[c] + xw[idx] * di * di;
}

// ---------------- edge scatter: one wave per edge, float4 per lane ----------------

__global__ __launch_bounds__(256) void gcn_scatter(
    const int* __restrict__ src, const int* __restrict__ dst,
    const float* __restrict__ dinv, const float* __restrict__ xw,
    float* __restrict__ out, int E)
{
    int e = (int)(((size_t)blockIdx.x * blockDim.x + threadIdx.x) >> 5);
    if (e >= E) return;
    int lane = threadIdx.x & 31;

    int s = src[e];
    int d = dst[e];
    float nrm = dinv[s] * dinv[d];

    const float4* row = (const float4*)(xw + (size_t)s * CH);
    float4 v = row[lane];                       // coalesced 512B row gather (L2 resident)

    float* o = out + (size_t)d * CH + lane * 4;
    atomicAdd(o + 0, v.x * nrm);
    atomicAdd(o + 1, v.y * nrm);
    atomicAdd(o + 2, v.z * nrm);
    atomicAdd(o + 3, v.w * nrm);
}

// ---------------- launch ----------------

extern "C" void kernel_launch(void* const* d_in, const int* in_sizes, int n_in,
                              void* d_out, int out_size, void* d_ws, size_t ws_size,
                              hipStream_t stream) {
    const float* x  = (const float*)d_in[0];
    const float* W  = (const float*)d_in[1];
    const float* b  = (const float*)d_in[2];
    const int*   ei = (const int*)d_in[3];

    const int n = in_sizes[0] / CH;   // nodes
    const int E = in_sizes[3] / 2;    // edges
    const int* src = ei;
    const int* dst = ei + E;

    float* deg = (float*)d_ws;        // n floats; becomes dinv in place
    float* xw  = deg + n;             // n*CH floats (offset n*4 bytes, 16B aligned for n%4==0)
    float* out = (float*)d_out;

    gcn_deg_init <<<(n + 255) / 256, 256, 0, stream>>>(deg, n);
    gcn_deg_accum<<<(E + 255) / 256, 256, 0, stream>>>(dst, deg, E);
    gcn_dinv     <<<(n + 255) / 256, 256, 0, stream>>>(deg, n);

    gcn_gemm_wmma<<<(n + 15) / 16, 256, 0, stream>>>(x, W, xw, n);

    int total = n * CH;
    gcn_init_out <<<(total + 255) / 256, 256, 0, stream>>>(xw, deg, b, out, total);

    size_t scatter_threads = (size_t)E * 32;
    gcn_scatter  <<<(int)((scatter_threads + 255) / 256), 256, 0, stream>>>(src, dst, deg, xw, out, E);
}